// Encoder_67396626808850
// MI455X (gfx1250) — compile-verified
//
#include <hip/hip_runtime.h>

// ---------------------------------------------------------------------------
// 2-layer GCN for MI455X (gfx1250, wave32).
//   layer: h = A_in @ W ; agg = scatter(dinv[s]*dinv[d]*h[s] -> d) + dinv^2*h + b
//          out = relu(agg)
// GEMMs run on the WMMA pipe (v_wmma_f32_16x16x4_f32, full fp32 precision —
// the workload is bound by the 2.5 GB of edge gather/scatter traffic, not by
// matrix math, so there is no reason to drop precision).
// ---------------------------------------------------------------------------

typedef __attribute__((ext_vector_type(2))) float v2f;
typedef __attribute__((ext_vector_type(8))) float v8f;

// ---------------- WMMA f32 GEMM: C[M,N] = A[M,K] x B[K,N] -------------------
// One wave32 per 16x16 output tile; V_WMMA_F32_16X16X4_F32, K stepped by 4.
// Requires M%16==0 (100000 = 6250*16), N%16==0, K%4==0.
//
// A 16x4 f32 fragment layout (ISA 7.12.2): lanes 0-15 -> rows 0-15, VGPR0=K0,
// VGPR1=K1; lanes 16-31 -> rows 0-15, K2/K3.  B 4x16: lane(0-15)=col, VGPRs
// hold K0/K1 (lanes 0-15) and K2/K3 (lanes 16-31).  C/D: VGPR r -> M=r
// (lanes 0-15) / M=r+8 (lanes 16-31), N=lane%16.
__global__ void gcn_gemm_wmma_f32(const float* __restrict__ A,
                                  const float* __restrict__ B,
                                  float* __restrict__ C,
                                  int M, int K, int N) {
  const int lane = threadIdx.x & 31;
  const int wid  = threadIdx.x >> 5;
  const int mtiles = M >> 4;
  const int ntiles = N >> 4;
  const int gw = blockIdx.x * (blockDim.x >> 5) + wid;
  if (gw >= mtiles * ntiles) return;
  const int tm = gw / ntiles;
  const int tn = gw - tm * ntiles;

  const int idx16 = lane & 15;          // A: row in tile; B/C: col in tile
  const int khalf = (lane >> 4) << 1;   // 0 (lanes 0-15) or 2 (lanes 16-31)

  const float* aRow = A + (size_t)(tm * 16 + idx16) * K + khalf;
  const float* bCol = B + (tn * 16 + idx16);

  v8f acc = {};
  for (int k = 0; k < K; k += 4) {
    v2f a = *(const v2f*)(aRow + k);          // A[row][k+khalf], A[row][k+khalf+1]
    v2f b;
    b.x = bCol[(size_t)(k + khalf) * N];      // B[k+khalf  ][col]
    b.y = bCol[(size_t)(k + khalf + 1) * N];  // B[k+khalf+1][col]
    acc = __builtin_amdgcn_wmma_f32_16x16x4_f32(
        /*neg_a=*/false, a, /*neg_b=*/false, b,
        /*c_mod=*/(short)0, acc, /*reuse_a=*/false, /*reuse_b=*/false);
  }

  const int mbase = tm * 16 + ((lane >> 4) << 3);
  float* cPtr = C + (size_t)mbase * N + tn * 16 + idx16;
#pragma unroll
  for (int r = 0; r < 8; ++r)
    cPtr[(size_t)r * N] = acc[r];
}

// ---------------- graph-normalization helpers -------------------------------
__global__ void gcn_zero_u32(unsigned* __restrict__ p, int n) {
  int i = blockIdx.x * blockDim.x + threadIdx.x;
  if (i < n) p[i] = 0u;
}

__global__ void gcn_degree(const int* __restrict__ dst, unsigned* __restrict__ deg,
                           int E) {
  int e = blockIdx.x * blockDim.x + threadIdx.x;
  if (e < E) atomicAdd(&deg[dst[e]], 1u);      // global_atomic_add_u32
}

__global__ void gcn_dinv(const unsigned* __restrict__ deg, float* __restrict__ dinv,
                         int n) {
  int i = blockIdx.x * blockDim.x + threadIdx.x;
  if (i < n) dinv[i] = rsqrtf((float)deg[i] + 1.0f);  // +1 = analytic self loop
}

// agg[i][c] = h[i][c] * dinv[i]^2 + b[c]   (also initializes agg buffer)
__global__ void gcn_selfloop_bias(const float4* __restrict__ h4,
                                  const float* __restrict__ dinv,
                                  const float4* __restrict__ b4,
                                  float4* __restrict__ agg4,
                                  int n, int Cq /* = C/4 */) {
  int gid = blockIdx.x * blockDim.x + threadIdx.x;
  if (gid >= n * Cq) return;
  int i = gid / Cq;
  int q = gid - i * Cq;
  float s  = dinv[i];
  float s2 = s * s;
  float4 h = h4[gid];
  float4 b = b4[q];
  float4 r;
  r.x = h.x * s2 + b.x;
  r.y = h.y * s2 + b.y;
  r.z = h.z * s2 + b.z;
  r.w = h.w * s2 + b.w;
  agg4[gid] = r;
}

// per-edge message: agg[dst] += dinv[src]*dinv[dst] * h[src]
// one thread per (edge, channel-quad); C=128 -> one wave per edge (coalesced
// 512B row gather), scatter via global_atomic_add_f32.
__global__ void gcn_edge_scatter(const float4* __restrict__ h4,
                                 const int* __restrict__ src,
                                 const int* __restrict__ dst,
                                 const float* __restrict__ dinv,
                                 float* __restrict__ agg,
                                 int E, int Cq /* = C/4 */) {
  int gid = blockIdx.x * blockDim.x + threadIdx.x;
  if (gid >= E * Cq) return;
  int e = gid / Cq;
  int q = gid - e * Cq;
  int s = src[e];
  int d = dst[e];
  float coef = dinv[s] * dinv[d];
  float4 v = h4[(size_t)s * Cq + q];
  float* p = agg + (size_t)d * (Cq * 4) + q * 4;
  atomicAdd(p + 0, v.x * coef);
  atomicAdd(p + 1, v.y * coef);
  atomicAdd(p + 2, v.z * coef);
  atomicAdd(p + 3, v.w * coef);
}

__global__ void gcn_relu4(float4* __restrict__ p, int n4) {
  int i = blockIdx.x * blockDim.x + threadIdx.x;
  if (i >= n4) return;
  float4 v = p[i];
  v.x = fmaxf(v.x, 0.0f);
  v.y = fmaxf(v.y, 0.0f);
  v.z = fmaxf(v.z, 0.0f);
  v.w = fmaxf(v.w, 0.0f);
  p[i] = v;
}

// ---------------------------------------------------------------------------
static inline size_t alignUp(size_t v, size_t a) { return (v + a - 1) & ~(a - 1); }
static inline int ceilDiv(int a, int b) { return (a + b - 1) / b; }

extern "C" void kernel_launch(void* const* d_in, const int* in_sizes, int n_in,
                              void* d_out, int out_size, void* d_ws, size_t ws_size,
                              hipStream_t stream) {
  (void)n_in; (void)out_size; (void)ws_size;

  const float* x  = (const float*)d_in[0];   // [N, IN]
  const int*   ei = (const int*)d_in[1];     // [2, E] (int32 per harness contract)
  const float* W1 = (const float*)d_in[2];   // [IN, HID]
  const float* b1 = (const float*)d_in[3];   // [HID]
  const float* W2 = (const float*)d_in[4];   // [HID, OUT]
  const float* b2 = (const float*)d_in[5];   // [OUT]
  float* out = (float*)d_out;                // [N, OUT]

  const int HID = in_sizes[3];               // 128
  const int OUT = in_sizes[5];               // 64
  const int E   = in_sizes[1] / 2;           // 1.6M
  const int IN  = in_sizes[2] / HID;         // 256
  const int N   = in_sizes[0] / IN;          // 100000 (multiple of 16)

  const int* src = ei;
  const int* dst = ei + E;

  // workspace layout
  char* ws = (char*)d_ws;
  size_t off = 0;
  unsigned* deg  = (unsigned*)(ws + off); off = alignUp(off + (size_t)N * 4, 256);
  float*    dinv = (float*)(ws + off);    off = alignUp(off + (size_t)N * 4, 256);
  float*    h1   = (float*)(ws + off);    off = alignUp(off + (size_t)N * HID * 4, 256);
  float*    agg1 = (float*)(ws + off);    off = alignUp(off + (size_t)N * HID * 4, 256);
  float*    h2   = (float*)(ws + off);    off = alignUp(off + (size_t)N * OUT * 4, 256);

  const int B = 256;

  // degree / normalization (shared by both layers)
  gcn_zero_u32<<<ceilDiv(N, B), B, 0, stream>>>(deg, N);
  gcn_degree<<<ceilDiv(E, B), B, 0, stream>>>(dst, deg, E);
  gcn_dinv<<<ceilDiv(N, B), B, 0, stream>>>(deg, dinv, N);

  // ---- layer 1: h1 = x @ W1 ; agg1 = norm-agg + bias ; relu in place ----
  {
    int tiles = (N >> 4) * (HID >> 4);
    gcn_gemm_wmma_f32<<<ceilDiv(tiles, B / 32), B, 0, stream>>>(x, W1, h1, N, IN, HID);
    int Cq = HID / 4;
    gcn_selfloop_bias<<<ceilDiv(N * Cq, B), B, 0, stream>>>(
        (const float4*)h1, dinv, (const float4*)b1, (float4*)agg1, N, Cq);
    gcn_edge_scatter<<<ceilDiv(E / 1 * Cq, B), B, 0, stream>>>(
        (const float4*)h1, src, dst, dinv, agg1, E, Cq);
    gcn_relu4<<<ceilDiv(N * Cq, B), B, 0, stream>>>((float4*)agg1, N * Cq);
  }

  // ---- layer 2: h2 = agg1 @ W2 ; out = norm-agg + bias ; relu in place ----
  {
    int tiles = (N >> 4) * (OUT >> 4);
    gcn_gemm_wmma_f32<<<ceilDiv(tiles, B / 32), B, 0, stream>>>(agg1, W2, h2, N, HID, OUT);
    int Cq = OUT / 4;
    gcn_selfloop_bias<<<ceilDiv(N * Cq, B), B, 0, stream>>>(
        (const float4*)h2, dinv, (const float4*)b2, (float4*)out, N, Cq);
    gcn_edge_scatter<<<ceilDiv(E * Cq, B), B, 0, stream>>>(
        (const float4*)h2, src, dst, dinv, out, E, Cq);
    gcn_relu4<<<ceilDiv(N * Cq, B), B, 0, stream>>>((float4*)out, N * Cq);
  }
}